// VectorMaxPool_7284264534094
// MI455X (gfx1250) — compile-verified
//
#include <hip/hip_runtime.h>

// Clang ext_vector types (accepted by nontemporal builtins, unlike HIP_vector_type)
typedef float f32x4 __attribute__((ext_vector_type(4)));
typedef float f32x2 __attribute__((ext_vector_type(2)));
typedef int   i32x4 __attribute__((ext_vector_type(4)));

#define AS1 __attribute__((address_space(1)))   // global ("__device__" in HIP diags)
#define AS3 __attribute__((address_space(3)))   // LDS

typedef AS1 i32x4 g_i32x4;   // global <4 x i32>
typedef AS3 i32x4 l_i32x4;   // LDS    <4 x i32>

// ---------------------------------------------------------------------------
// Problem constants (B=16, C=64, H=W=256, K=2 pool)
// Output rows R = B*C*Ho = 131072; output row r consumes input [r*512, r*512+512)
// (two contiguous 256-float input rows; Ho rows tile H contiguously per (b,c)).
// ---------------------------------------------------------------------------
#define OUT_ROWS    131072
#define IN_ROW_ELTS 512
#define W_IN        256
#define WO          128
#define N_PAIRS     (OUT_ROWS * 64)       // 8,388,608 work items (2 out pixels each)
#define OUT_ELEMS   (OUT_ROWS * WO)       // 16,777,216 floats per output field

#define BLOCK   256                        // 8 wave32s
#define ITER    8                          // chunks per block (double-buffered)
#define PER_BLOCK (BLOCK * ITER)           // 2048 work items per block

// LDS: 2 buffers x 4 streams x 256 lanes x 16B = 32 KB
#define BUF_BYTES    16384
#define STREAM_BYTES 4096

// First strict max of squared magnitude over window order
// (r0,c0),(r0,c1),(r1,c0),(r1,c1)  == patch index dh*2+dw (first-max tie-break).
__device__ __forceinline__ void pick4(float ua, float va, float ub, float vb,
                                      float uc, float vc, float ud, float vd,
                                      float& uo, float& vo) {
    float ma = fmaf(ua, ua, va * va);
    float mb = fmaf(ub, ub, vb * vb);
    float mc = fmaf(uc, uc, vc * vc);
    float md = fmaf(ud, ud, vd * vd);
    float bu = ua, bv = va, bm = ma;
    if (mb > bm) { bm = mb; bu = ub; bv = vb; }
    if (mc > bm) { bm = mc; bu = uc; bv = vc; }
    if (md > bm) { bm = md; bu = ud; bv = vd; }
    uo = bu; vo = bv;
}

__global__ __launch_bounds__(BLOCK)
void VectorMaxPool_7284264534094_kernel(const float* __restrict__ u,
                                        const float* __restrict__ v,
                                        float* __restrict__ uo,
                                        float* __restrict__ vo) {
    __shared__ __align__(16) char smem[2 * BUF_BYTES];

    const unsigned tid  = threadIdx.x;
    const unsigned base = blockIdx.x * PER_BLOCK;

    // Stage the 4 b128 segments for work item q into LDS buffer `buf`
    // via the gfx1250 async-to-LDS engine (tracked on ASYNCcnt).
    auto issue4 = [&](unsigned buf, unsigned q) {
        const unsigned p  = q & 63u;        // pair-of-outputs within the row
        const unsigned r  = q >> 6;         // fused (b,c,ho) output row id
        const size_t in0  = (size_t)r * IN_ROW_ELTS + (size_t)p * 4;  // input row 0
        char* lb = smem + buf * BUF_BYTES + tid * 16;
        __builtin_amdgcn_global_load_async_to_lds_b128(
            (g_i32x4*)(u + in0),         (l_i32x4*)(lb + 0 * STREAM_BYTES), 0, 0);
        __builtin_amdgcn_global_load_async_to_lds_b128(
            (g_i32x4*)(u + in0 + W_IN),  (l_i32x4*)(lb + 1 * STREAM_BYTES), 0, 0);
        __builtin_amdgcn_global_load_async_to_lds_b128(
            (g_i32x4*)(v + in0),         (l_i32x4*)(lb + 2 * STREAM_BYTES), 0, 0);
        __builtin_amdgcn_global_load_async_to_lds_b128(
            (g_i32x4*)(v + in0 + W_IN),  (l_i32x4*)(lb + 3 * STREAM_BYTES), 0, 0);
    };

    issue4(0, base + tid);                               // prologue: chunk 0

    for (int i = 0; i < ITER; ++i) {
        if (i + 1 < ITER) {
            issue4((unsigned)(i + 1) & 1u, base + (unsigned)(i + 1) * BLOCK + tid);
            __builtin_amdgcn_s_wait_asynccnt(4);         // chunk i landed (in-order)
        } else {
            __builtin_amdgcn_s_wait_asynccnt(0);         // drain
        }

        // Conflict-free ds_load_b128 reads (lane stride 16B within each stream).
        const char* lb = smem + ((unsigned)i & 1u) * BUF_BYTES + tid * 16;
        const f32x4 u0 = *(const f32x4*)(lb + 0 * STREAM_BYTES);
        const f32x4 u1 = *(const f32x4*)(lb + 1 * STREAM_BYTES);
        const f32x4 v0 = *(const f32x4*)(lb + 2 * STREAM_BYTES);
        const f32x4 v1 = *(const f32x4*)(lb + 3 * STREAM_BYTES);

        float uo0, vo0, uo1, vo1;
        pick4(u0.x, v0.x, u0.y, v0.y, u1.x, v1.x, u1.y, v1.y, uo0, vo0);
        pick4(u0.z, v0.z, u0.w, v0.w, u1.z, v1.z, u1.w, v1.w, uo1, vo1);

        const unsigned q = base + (unsigned)i * BLOCK + tid;
        const unsigned p = q & 63u;
        const unsigned r = q >> 6;
        const size_t o = (size_t)r * WO + (size_t)p * 2;

        f32x2 up; up.x = uo0; up.y = uo1;
        f32x2 vp; vp.x = vo0; vp.y = vo1;
        __builtin_nontemporal_store(up, (f32x2*)(uo + o));   // streamed once: TH=NT
        __builtin_nontemporal_store(vp, (f32x2*)(vo + o));
    }
}

extern "C" void kernel_launch(void* const* d_in, const int* in_sizes, int n_in,
                              void* d_out, int out_size, void* d_ws, size_t ws_size,
                              hipStream_t stream) {
    (void)in_sizes; (void)n_in; (void)out_size; (void)d_ws; (void)ws_size;
    const float* u = (const float*)d_in[0];
    const float* v = (const float*)d_in[1];
    float* uo = (float*)d_out;               // u_out, 16,777,216 floats
    float* vo = (float*)d_out + OUT_ELEMS;   // v_out, concatenated flat

    const int grid = N_PAIRS / PER_BLOCK;    // 4096 blocks, exact division
    VectorMaxPool_7284264534094_kernel<<<grid, BLOCK, 0, stream>>>(u, v, uo, vo);
}